// Encoder1DCNN_LSTM_21251498181453
// MI455X (gfx1250) — compile-verified
//
#include <hip/hip_runtime.h>
#include <hip/hip_bf16.h>
#include <math.h>

typedef float v2f __attribute__((ext_vector_type(2)));
typedef float v8f __attribute__((ext_vector_type(8)));

#define SELU_SCALE 1.0507009873554805f
#define SELU_ALPHA 1.6732632423543772f

// Hardware tanh (CDNA5 V_TANH_F32 transcendental).
__device__ __forceinline__ float fast_tanh(float v) {
#if __has_builtin(__builtin_amdgcn_tanhf)
  return __builtin_amdgcn_tanhf(v);
#else
  float r;
  asm("v_tanh_f32 %0, %1\n\tv_nop" : "=v"(r) : "v"(v));
  return r;
#endif
}

// ---------------------------------------------------------------------------
// Conv1d (K=5, VALID) + optional SELU.
// Input layout: in_tbc ? (T_in, B, Cin) : (B, Cin, L_in)   [stage 0 uses NCH]
// Output layout: out_nch ? (B, Cout, T) : (T, B, Cout)
// ---------------------------------------------------------------------------
__global__ void conv1d_selu_kernel(const float* __restrict__ in,
                                   const float* __restrict__ w,
                                   const float* __restrict__ bias,
                                   float* __restrict__ out,
                                   int B, int Cin, int Cout, int Lin, int Tout,
                                   int in_tbc, int out_nch, int do_selu) {
  int idx = blockIdx.x * blockDim.x + threadIdx.x;
  int total = B * Cout * Tout;
  if (idx >= total) return;
  int co = idx % Cout;
  int b  = (idx / Cout) % B;
  int t  = idx / (Cout * B);
  float acc = bias[co];
  for (int ci = 0; ci < Cin; ++ci) {
    #pragma unroll
    for (int k = 0; k < 5; ++k) {
      float xv = in_tbc ? in[((size_t)(t + k) * B + b) * Cin + ci]
                        : in[((size_t)b * Cin + ci) * Lin + (t + k)];
      acc += xv * w[(co * Cin + ci) * 5 + k];
    }
  }
  if (do_selu)
    acc = acc > 0.f ? SELU_SCALE * acc
                    : SELU_SCALE * SELU_ALPHA * (__expf(acc) - 1.f);
  if (out_nch) out[((size_t)b * Cout + co) * Tout + t] = acc;
  else         out[((size_t)t * B + b) * Cout + co] = acc;
}

// ---------------------------------------------------------------------------
// Pack LSTM weights, PAIR-INTERLEAVED for single-b64 B-fragment loads:
//   dst[((k>>1)*NP + p)*2 + (k&1)] = W[gate_row(p)][k]
//   padded column p: tile j=p/16, gate=j&3, seg s=j>>2, h = s*16 + (p&15)
// Rows padded to KR = 4*ceil(H/4) with zeros; invalid h columns zeroed.
// ---------------------------------------------------------------------------
__global__ void pack_lstm_kernel(const float* __restrict__ wih,
                                 const float* __restrict__ whh,
                                 const float* __restrict__ bih,
                                 const float* __restrict__ bhh,
                                 float* __restrict__ wihP,
                                 float* __restrict__ whhP,
                                 float* __restrict__ biasP,
                                 int H, int KR, int NP) {
  int idx = blockIdx.x * blockDim.x + threadIdx.x;
  if (idx >= KR * NP) return;
  int k = idx / NP, p = idx % NP;
  int j = p >> 4, n = p & 15;
  int gate = j & 3, s = j >> 2;
  int h = s * 16 + n;
  bool valid = (h < H) && (k < H);
  int row = gate * H + h;
  int dst = (((k >> 1) * NP + p) << 1) + (k & 1);
  wihP[dst] = valid ? wih[row * H + k] : 0.f;
  whhP[dst] = valid ? whh[row * H + k] : 0.f;
  if (k == 0) biasP[p] = (h < H) ? (bih[row] + bhh[row]) : 0.f;
}

// ---------------------------------------------------------------------------
// LSTM scan: one wave (32 threads) per 16 batch rows, time loop inside.
// Gate matrix G(16 x 4H padded) = x_t * WihP + h * WhhP via
// V_WMMA_F32_16X16X4_F32 starting from the inline-0 C operand; bias folded
// into the nonlinearities. Gate tiles packed so tile 4s+{0,1,2,3}={i,f,g,o}
// for h-range [16s,16s+16) -> cell update is lane-aligned with c fragments.
// No k<H guards: weight rows k>=H are zero, and every over-read location
// holds finite data (finite * 0 == 0), so fragments load unconditionally.
// x layout: (T, B, H). out layout: (T, B, H). x_t loads pipelined one step.
// ---------------------------------------------------------------------------
template <int H>
__global__ void lstm_scan_kernel(const float* __restrict__ x,
                                 const float* __restrict__ wihP,
                                 const float* __restrict__ whhP,
                                 const float* __restrict__ biasP,
                                 const float* __restrict__ h0,
                                 const float* __restrict__ c0,
                                 float* __restrict__ out,
                                 int T, int Bsz) {
  constexpr int NTPG = (H + 15) / 16;   // 16-wide segments per gate
  constexpr int NT   = 4 * NTPG;        // total N tiles
  constexpr int KC   = (H + 3) / 4;     // K chunks of 4
  constexpr int KR   = 4 * KC;          // padded K rows
  constexpr int NP   = NT * 16;         // padded N

  __shared__ __align__(16) float s_wih[KR * NP];  // pair-interleaved
  __shared__ __align__(16) float s_h[16 * 36];    // h transpose buffer

  const int lane  = threadIdx.x;        // 0..31
  const int b0    = blockIdx.x * 16;
  const int nloc  = lane & 15;          // N within tile / M within A-frag
  const int khalf = (lane >> 4) * 2;    // A/B frag K sub-offset (0 or 2)
  const int kpair = (lane >> 4);        // pair index offset (k0>>1 low bit)
  const int mrow  = (lane >> 4) * 8;    // C/D frag M base (0 or 8)

  for (int i = lane; i < KR * NP; i += 32) s_wih[i] = wihP[i];
  __syncthreads();

  // Preload recurrent weight B-fragments (one b64 load each).
  v2f wB[KC][NT];
  #pragma unroll
  for (int kc = 0; kc < KC; ++kc) {
    #pragma unroll
    for (int j = 0; j < NT; ++j) {
      int p = j * 16 + nloc;
      wB[kc][j] = *(const v2f*)&whhP[(((size_t)(2 * kc + kpair) * NP) + p) * 2];
    }
  }
  // Bias: per lane per tile. hb = 0.5*b fuses into the sigmoid pre-scale.
  float bv[NT], hb[NT];
  #pragma unroll
  for (int j = 0; j < NT; ++j) {
    bv[j] = biasP[j * 16 + nloc];
    hb[j] = 0.5f * bv[j];
  }

  // Initial states in C/D fragment layout.
  v8f cfrag[NTPG], hfrag[NTPG];
  #pragma unroll
  for (int s = 0; s < NTPG; ++s) {
    #pragma unroll
    for (int r = 0; r < 8; ++r) {
      int m = mrow + r, h = s * 16 + nloc;
      cfrag[s][r] = (h < H) ? c0[(size_t)(b0 + m) * H + h] : 0.f;
      hfrag[s][r] = (h < H) ? h0[(size_t)(b0 + m) * H + h] : 0.f;
    }
  }

  auto load_xA = [&](int t, v2f* xA) {
    const float* xt = x + (size_t)t * Bsz * H + (size_t)(b0 + nloc) * H;
    #pragma unroll
    for (int kc = 0; kc < KC; ++kc) {
      int k0 = 4 * kc + khalf;
      xA[kc].x = xt[k0];      // k >= H reads finite data; weight row is 0
      xA[kc].y = xt[k0 + 1];
    }
  };

  v2f xA[KC], xAn[KC];
  load_xA(0, xA);

  for (int t = 0; t < T; ++t) {
    // Prefetch next step's x fragments (overlaps WMMA chain below).
    if (t + 1 < T) load_xA(t + 1, xAn);

    // h: D-layout -> LDS -> A-layout (DS ops in-order within a wave).
    #pragma unroll
    for (int s = 0; s < NTPG; ++s)
      #pragma unroll
      for (int r = 0; r < 8; ++r)
        s_h[(mrow + r) * 36 + s * 16 + nloc] = hfrag[s][r];
    asm volatile("s_wait_dscnt 0x0" ::: "memory");

    // hA: contiguous k0,k0+1 (k0 even) -> single aligned ds_load_b64.
    v2f hA[KC];
    #pragma unroll
    for (int kc = 0; kc < KC; ++kc)
      hA[kc] = *(const v2f*)&s_h[nloc * 36 + 4 * kc + khalf];

    // Gate accumulation: x*Wih + h*Whh, starting from inline-0 C operand.
    v8f acc[NT];
    #pragma unroll
    for (int kc = 0; kc < KC; ++kc) {
      #pragma unroll
      for (int j = 0; j < NT; ++j) {
        v2f bI = *(const v2f*)&s_wih[((2 * kc + kpair) * NP + j * 16 + nloc) * 2];
        if (kc == 0) {
          v8f z = {};  // lowers to the inline 0 SRC2 operand: no init movs
          acc[j] = __builtin_amdgcn_wmma_f32_16x16x4_f32(
              false, xA[kc], false, bI, (short)0, z, false, false);
        } else {
          acc[j] = __builtin_amdgcn_wmma_f32_16x16x4_f32(
              false, xA[kc], false, bI, (short)0, acc[j], false, false);
        }
        acc[j] = __builtin_amdgcn_wmma_f32_16x16x4_f32(
            false, hA[kc], false, wB[kc][j], (short)0, acc[j], false, false);
      }
    }

    // Cell update: tiles 4s+{0,1,2,3} = {i,f,g,o}, lane-aligned with cfrag[s].
    // sigmoid(g+b) = 0.5*tanh(0.5*g + 0.5*b) + 0.5  (bias fused into the fma).
    #pragma unroll
    for (int s = 0; s < NTPG; ++s) {
      #pragma unroll
      for (int r = 0; r < 8; ++r) {
        float iv = fmaf(0.5f, fast_tanh(fmaf(0.5f, acc[4 * s + 0][r], hb[4 * s + 0])), 0.5f);
        float fv = fmaf(0.5f, fast_tanh(fmaf(0.5f, acc[4 * s + 1][r], hb[4 * s + 1])), 0.5f);
        float gv = fast_tanh(acc[4 * s + 2][r] + bv[4 * s + 2]);
        float ov = fmaf(0.5f, fast_tanh(fmaf(0.5f, acc[4 * s + 3][r], hb[4 * s + 3])), 0.5f);
        float cv = fmaf(fv, cfrag[s][r], iv * gv);
        cfrag[s][r] = cv;
        hfrag[s][r] = ov * fast_tanh(cv);
      }
    }

    // Emit h_t to (T,B,H): lanes 0-15 / 16-31 write contiguous 64B runs.
    float* ot = out + (size_t)t * Bsz * H;
    #pragma unroll
    for (int s = 0; s < NTPG; ++s) {
      int h = s * 16 + nloc;
      if (h < H) {
        #pragma unroll
        for (int r = 0; r < 8; ++r)
          ot[(size_t)(b0 + mrow + r) * H + h] = hfrag[s][r];
      }
    }

    // Rotate pipelined x fragments.
    #pragma unroll
    for (int kc = 0; kc < KC; ++kc) xA[kc] = xAn[kc];
  }
}

// ---------------------------------------------------------------------------
extern "C" void kernel_launch(void* const* d_in, const int* in_sizes, int n_in,
                              void* d_out, int out_size, void* d_ws, size_t ws_size,
                              hipStream_t stream) {
  (void)in_sizes; (void)n_in; (void)out_size; (void)ws_size;
  const int B = 1024;
  const int Hs[5] = {5, 10, 20, 20, 20};

  // Workspace: S0 (conv out, T-major), S1 (lstm out, T-major), packed weights.
  const size_t BUFSZ = (size_t)20736000;  // >= 1012*1024*20 floats (+ slack)
  float* S0    = (float*)d_ws;
  float* S1    = S0 + BUFSZ;
  float* wihP  = S1 + BUFSZ;
  float* whhP  = wihP + 4096;             // KR*NP <= 20*128 = 2560
  float* biasP = whhP + 4096;

  const float* cur = (const float*)d_in[0];   // (B, 2, 1024) NCH
  int in_tbc = 0;
  int L = 1024;
  int Cin = 2;

  for (int i = 0; i < 5; ++i) {
    const int H = Hs[i];
    const int T = L - 4;
    const float* cw = (const float*)d_in[1 + i];
    const float* cb = (const float*)d_in[7 + i];
    const float* wih = (const float*)d_in[13 + 4 * i];
    const float* whh = (const float*)d_in[14 + 4 * i];
    const float* bih = (const float*)d_in[15 + 4 * i];
    const float* bhh = (const float*)d_in[16 + 4 * i];
    const float* h0  = (const float*)d_in[33 + i];
    const float* c0  = (const float*)d_in[38 + i];

    int total = B * H * T;
    conv1d_selu_kernel<<<(total + 255) / 256, 256, 0, stream>>>(
        cur, cw, cb, S0, B, Cin, H, L, T, in_tbc, /*out_nch=*/0, /*selu=*/1);

    int KC = (H + 3) / 4, KR = 4 * KC;
    int NTPG = (H + 15) / 16, NP = 4 * NTPG * 16;
    int ptotal = KR * NP;
    pack_lstm_kernel<<<(ptotal + 127) / 128, 128, 0, stream>>>(
        wih, whh, bih, bhh, wihP, whhP, biasP, H, KR, NP);

    if (H == 5)
      lstm_scan_kernel<5><<<B / 16, 32, 0, stream>>>(S0, wihP, whhP, biasP, h0, c0, S1, T, B);
    else if (H == 10)
      lstm_scan_kernel<10><<<B / 16, 32, 0, stream>>>(S0, wihP, whhP, biasP, h0, c0, S1, T, B);
    else
      lstm_scan_kernel<20><<<B / 16, 32, 0, stream>>>(S0, wihP, whhP, biasP, h0, c0, S1, T, B);

    cur = S1;
    in_tbc = 1;
    L = T;
    Cin = H;
  }

  // conv_last: (T=1004,B,20) -> d_out (B,10,1000) NCH, no SELU.
  const int Tf = L - 4;  // 1000
  int total = B * 10 * Tf;
  conv1d_selu_kernel<<<(total + 255) / 256, 256, 0, stream>>>(
      cur, (const float*)d_in[6], (const float*)d_in[12], (float*)d_out,
      B, 20, 10, L, Tf, /*in_tbc=*/1, /*out_nch=*/1, /*selu=*/0);
}